// ExponentialMovingAverage_68436008894477
// MI455X (gfx1250) — compile-verified
//
#include <hip/hip_runtime.h>

// ---------------------------------------------------------------------------
// EMA scan via WMMA f32 16x16x4 tile-scan, chunk-parallel (decay-truncated).
//   y_t = ALPHA*x_t + BETA*y_{t-1},  y_0 = x_0
// Shapes: x[16][4096][512] f32 -> out same.
// ---------------------------------------------------------------------------

typedef __attribute__((ext_vector_type(2))) float v2f;
typedef __attribute__((ext_vector_type(8))) float v8f;

#define EMA_ALPHA 0.1f
#define EMA_BETA  0.9f

#define BATCH   16
#define TLEN    4096
#define CHAN    512
#define LCHUNK  512                     // timesteps per chunk
#define NCHUNK  (TLEN / LCHUNK)         // 8
#define WARM    256                     // phase-1 warm-up (beta^256 ~ 2e-12)
#define CGROUPS (CHAN / 16)             // 32 channel groups of 16
#define WAVES_TOTAL (BATCH * NCHUNK * CGROUPS)  // 4096 waves
#define BLOCK_THREADS 256
#define WAVES_PER_BLOCK (BLOCK_THREADS / 32)
#define GRID_BLOCKS (WAVES_TOTAL / WAVES_PER_BLOCK)  // 512

// W[m][k] = (k<=m) ? ALPHA * BETA^(m-k) : 0   (16x16 lower-triangular scan matrix)
__device__ __forceinline__ float wcoef(int m, int k) {
  if (k > m) return 0.0f;
  float p = EMA_ALPHA;
  for (int i = 0; i < m - k; ++i) p *= EMA_BETA;
  return p;
}

struct WaveCtx {
  v2f a[4];   // A-matrix (W) registers for the 4 chained K=4 WMMAs
  v8f d;      // d[v] = BETA^(v + 8h + 1): carry-injection coefficients (C = d * carry)
  int m;      // lane & 15 : A-row / B-col / C,D-col
  int h;      // lane >> 4 : half-wave selector (K / M split)
};

__device__ __forceinline__ WaveCtx make_ctx(int lane) {
  WaveCtx c;
  c.m = lane & 15;
  c.h = lane >> 4;
  // A layout (f32 16x4, MxK): VGPR0 = K=2h, VGPR1 = K=2h+1 for lane-half h.
#pragma unroll
  for (int q = 0; q < 4; ++q) {
    int k = 4 * q + 2 * c.h;
    c.a[q].x = wcoef(c.m, k);
    c.a[q].y = wcoef(c.m, k + 1);
  }
  // d[v] = BETA^(v + 8h + 1); C/D lane holds rows M = v + 8h.
  float p = EMA_BETA;
  for (int i = 0; i < 8 * c.h; ++i) p *= EMA_BETA;
#pragma unroll
  for (int v = 0; v < 8; ++v) { c.d[v] = p; p *= EMA_BETA; }
  return c;
}

// One 16-timestep x 16-channel tile: acc(D) = W * X + acc(C).
// xt points at x[t0][c0]; row stride CHAN floats.
// scale_t0: fold y_0 = x_0 by feeding 10*x_0 (W[0][0] = ALPHA -> ALPHA*10*x0 ~= x0).
__device__ __forceinline__ v8f ema_tile(const float* __restrict__ xt,
                                        const WaveCtx& c, v8f acc, bool scale_t0) {
#pragma unroll
  for (int q = 0; q < 4; ++q) {
    int r = 4 * q + 2 * c.h;               // B rows (K) for this lane-half
    float bx = xt[(size_t)r * CHAN + c.m];
    float by = xt[(size_t)(r + 1) * CHAN + c.m];
    bx = (scale_t0 && r == 0) ? bx * 10.0f : bx;   // pure v_cndmask, EXEC untouched
    v2f bv; bv.x = bx; bv.y = by;
    acc = __builtin_amdgcn_wmma_f32_16x16x4_f32(
        /*neg_a=*/false, c.a[q], /*neg_b=*/false, bv,
        /*c_mod=*/(short)0, acc, /*reuse_a=*/false, /*reuse_b=*/false);
  }
  return acc;
}

// Broadcast new carry (D row 15 = acc[7] of lanes 16..31) to all lanes.
__device__ __forceinline__ float carry_bcast(v8f acc, int m) {
  return __shfl(acc[7], 16 + m, 32);
}

// Phase 1: per (b, chunk, cgroup) wave, scan the LAST `WARM` steps of the chunk
// with zero carry; write the end value (the chunk's outgoing carry, exact to
// within beta^WARM ~ 2e-12 rel.) to workspace.
__global__ void __launch_bounds__(BLOCK_THREADS)
ema_phase1(const float* __restrict__ x, float* __restrict__ ecarry) {
  const int lane = threadIdx.x & 31;
  const int wave = blockIdx.x * WAVES_PER_BLOCK + (threadIdx.x >> 5);
  const int cg = wave % CGROUPS;
  const int j  = (wave / CGROUPS) % NCHUNK;
  const int b  = wave / (CGROUPS * NCHUNK);
  const WaveCtx c = make_ctx(lane);

  const int t0 = j * LCHUNK + (LCHUNK - WARM);
  const float* base = x + ((size_t)b * TLEN + t0) * CHAN + (size_t)cg * 16;

  v8f acc = {};                 // zero carry-in
  float carry = 0.0f;
  for (int tile = 0; tile < WARM / 16; ++tile) {
    const float* xt = base + (size_t)tile * 16 * CHAN;
    if (tile + 1 < WARM / 16) __builtin_prefetch(xt + 16 * CHAN, 0, 0);
    acc = ema_tile(xt, c, acc, false);
    carry = carry_bcast(acc, c.m);
#pragma unroll
    for (int v = 0; v < 8; ++v) acc[v] = c.d[v] * carry;
  }
  if (lane < 16)
    ecarry[((size_t)b * NCHUNK + j) * CHAN + (size_t)cg * 16 + lane] = carry;
}

// Phase 3: full scan of each chunk seeded with the previous chunk's carry
// (beta^WARM ~ 2e-12 makes that carry exact in fp32); write final output.
__global__ void __launch_bounds__(BLOCK_THREADS)
ema_phase3(const float* __restrict__ x, const float* __restrict__ ecarry,
           float* __restrict__ out) {
  const int lane = threadIdx.x & 31;
  const int wave = blockIdx.x * WAVES_PER_BLOCK + (threadIdx.x >> 5);
  const int cg = wave % CGROUPS;
  const int j  = (wave / CGROUPS) % NCHUNK;
  const int b  = wave / (CGROUPS * NCHUNK);
  const WaveCtx c = make_ctx(lane);

  float carry = 0.0f;
  if (j > 0)
    carry = ecarry[((size_t)b * NCHUNK + (j - 1)) * CHAN + (size_t)cg * 16 + c.m];

  v8f acc;
#pragma unroll
  for (int v = 0; v < 8; ++v) acc[v] = c.d[v] * carry;

  const int t0 = j * LCHUNK;
  const float* xb = x   + ((size_t)b * TLEN + t0) * CHAN + (size_t)cg * 16;
  float*       ob = out + ((size_t)b * TLEN + t0) * CHAN + (size_t)cg * 16;

  for (int tile = 0; tile < LCHUNK / 16; ++tile) {
    const float* xt = xb + (size_t)tile * 16 * CHAN;
    if (tile + 1 < LCHUNK / 16) __builtin_prefetch(xt + 16 * CHAN, 0, 0);
    const bool sc = (j == 0) && (tile == 0);   // t = 0 special case
    acc = ema_tile(xt, c, acc, sc);

    float* ot = ob + (size_t)tile * 16 * CHAN;
#pragma unroll
    for (int v = 0; v < 8; ++v)
      ot[(size_t)(v + 8 * c.h) * CHAN + c.m] = acc[v];

    float cv = carry_bcast(acc, c.m);
#pragma unroll
    for (int v = 0; v < 8; ++v) acc[v] = c.d[v] * cv;
  }
}

extern "C" void kernel_launch(void* const* d_in, const int* in_sizes, int n_in,
                              void* d_out, int out_size, void* d_ws, size_t ws_size,
                              hipStream_t stream) {
  (void)in_sizes; (void)n_in; (void)out_size; (void)ws_size;
  const float* x = (const float*)d_in[0];
  float* out = (float*)d_out;
  float* ecarry = (float*)d_ws;   // BATCH*NCHUNK*CHAN floats = 256 KB

  ema_phase1<<<GRID_BLOCKS, BLOCK_THREADS, 0, stream>>>(x, ecarry);
  ema_phase3<<<GRID_BLOCKS, BLOCK_THREADS, 0, stream>>>(x, ecarry, out);
}